// ConcatLayer_3556232921567
// MI455X (gfx1250) — compile-verified
//
#include <hip/hip_runtime.h>
#include <stdint.h>

// ---------------------------------------------------------------------------
// Row-wise select kernel for the (N x 9) -> (N x 3) reference.
// Memory-bound streaming problem: 192 MB total traffic -> ~8.2 us floor at
// 23.3 TB/s. Strategy: tile 1024 rows/block through LDS using CDNA5 async
// DMA (GLOBAL_LOAD_ASYNC_TO_LDS_B128 / GLOBAL_STORE_ASYNC_FROM_LDS_B128,
// tracked by ASYNCcnt), so every global transaction is a full-width,
// lane-consecutive b128 despite the awkward 36-byte row pitch.
// ---------------------------------------------------------------------------

#define BLOCK     256
#define TILE_ROWS 1024
#define IN_F4     (TILE_ROWS * 9 / 4)   // 2304 float4 per tile (input)
#define OUT_F4    (TILE_ROWS * 3 / 4)   //  768 float4 per tile (output)

// _max_index: tie-for-max -> 0, else 1 - argmax (first-occurrence argmax).
__device__ __forceinline__ int max_index3(float a, float b, float c) {
    float m  = fmaxf(a, fmaxf(b, c));
    int   ea = (a == m), eb = (b == m), ec = (c == m);
    int   cnt = ea + eb + ec;
    int   arg = ea ? 0 : (eb ? 1 : 2);   // first max
    return (cnt > 1) ? 0 : (1 - arg);    // in {-1, 0, 1}
}

__device__ __forceinline__ float sel3(float a, float b, float c, int i) {
    return (i == 0) ? a : ((i == 1) ? b : c);
}

__device__ __forceinline__ void compute_row(const float* __restrict__ v,
                                            float* __restrict__ o) {
    const int mu = max_index3(v[0], v[1], v[2]);
    const int mn = max_index3(v[3], v[4], v[5]);
    const int md = max_index3(v[6], v[7], v[8]);

    // calc = |mn| * (mu + mn + md);  |mn| in {0,1}
    const int calc = (mn == 0) ? 0 : (mu + mn + md);
    const int sc   = (calc > 0) - (calc < 0);          // sign(calc)

    // keep(v, m): zero the slice unless sign(m) == sc  (sign(m)==m here)
    const float u0 = (mu == sc) ? v[0] : 0.f;
    const float u1 = (mu == sc) ? v[1] : 0.f;
    const float u2 = (mu == sc) ? v[2] : 0.f;
    const float n0 = (mn == sc) ? v[3] : 0.f;
    const float n1 = (mn == sc) ? v[4] : 0.f;
    const float n2 = (mn == sc) ? v[5] : 0.f;
    const float d0 = (md == sc) ? v[6] : 0.f;
    const float d1 = (md == sc) ? v[7] : 0.f;
    const float d2 = (md == sc) ? v[8] : 0.f;

    // idx = calc==0 ? 1 : (calc>=1 ? 0 : 2)
    const int idx = (calc == 0) ? 1 : ((calc > 0) ? 0 : 2);

    const float t0 = sel3(u0, u1, u2, idx);
    const float t1 = sel3(n0, n1, n2, idx);
    const float t2 = sel3(d0, d1, d2, idx);

    // first-occurrence argmax over (t0, t1, t2)
    const int choice = (t0 >= t1 && t0 >= t2) ? 0 : ((t1 >= t2) ? 1 : 2);

    o[0] = sel3(u0, n0, d0, choice);
    o[1] = sel3(u1, n1, d1, choice);
    o[2] = sel3(u2, n2, d2, choice);
}

__global__ void __launch_bounds__(BLOCK)
concat_layer_kernel(const float* __restrict__ x, float* __restrict__ out,
                    int n_rows, int full_tiles) {
    __shared__ float4 smem[IN_F4 + OUT_F4];     // 36 KB + 12 KB = 48 KB
    const int t = threadIdx.x;
    const int b = blockIdx.x;

    float*       sIn  = (float*)smem;
    float*       sOut = (float*)(smem + IN_F4);

    if (b < full_tiles) {
        // Flat->LDS mapping uses addr[31:0], so truncating the generic
        // pointer yields the LDS byte offset the async ops require.
        const uint32_t lds_in  = (uint32_t)(uintptr_t)(smem);
        const uint32_t lds_out = (uint32_t)(uintptr_t)(smem + IN_F4);
        const uint64_t gin  = (uint64_t)(uintptr_t)x   + (uint64_t)b * (TILE_ROWS * 36);
        const uint64_t gout = (uint64_t)(uintptr_t)out + (uint64_t)b * (TILE_ROWS * 12);

        // --- Stage 1: async DMA global -> LDS (9 x b128 per thread) -------
        #pragma unroll
        for (int k = 0; k < 9; ++k) {
            const uint32_t f4 = (uint32_t)(t + k * BLOCK);
            const uint64_t ga = gin + (uint64_t)f4 * 16u;
            const uint32_t la = lds_in + f4 * 16u;
            asm volatile("global_load_async_to_lds_b128 %0, %1, off"
                         :: "v"(la), "v"(ga) : "memory");
        }
        asm volatile("s_wait_asynccnt 0" ::: "memory");   // own ASYNCcnt -> 0
        __syncthreads();                                  // all waves' tiles landed

        // --- Stage 2: compute 4 rows/thread from LDS ----------------------
        // read stride 9 words, write stride 3 words: both conflict-free (gcd with 64 banks = 1)
        #pragma unroll
        for (int j = 0; j < TILE_ROWS / BLOCK; ++j) {
            const int r = t + j * BLOCK;
            float o[3];
            compute_row(sIn + r * 9, o);
            sOut[r * 3 + 0] = o[0];
            sOut[r * 3 + 1] = o[1];
            sOut[r * 3 + 2] = o[2];
        }
        __syncthreads();                                  // LDS results visible

        // --- Stage 3: async DMA LDS -> global (3 x b128 per thread) -------
        #pragma unroll
        for (int k = 0; k < 3; ++k) {
            const uint32_t f4 = (uint32_t)(t + k * BLOCK);
            const uint64_t ga = gout + (uint64_t)f4 * 16u;
            const uint32_t la = lds_out + f4 * 16u;
            asm volatile("global_store_async_from_lds_b128 %0, %1, off"
                         :: "v"(ga), "v"(la) : "memory");
        }
        asm volatile("s_wait_asynccnt 0" ::: "memory");   // stores drained before LDS dies
    } else {
        // Tail block: direct guarded path (one block; perf-irrelevant).
        const int base = full_tiles * TILE_ROWS;
        for (int r = base + t; r < n_rows; r += BLOCK) {
            float v[9];
            #pragma unroll
            for (int e = 0; e < 9; ++e) v[e] = x[(size_t)r * 9 + e];
            float o[3];
            compute_row(v, o);
            out[(size_t)r * 3 + 0] = o[0];
            out[(size_t)r * 3 + 1] = o[1];
            out[(size_t)r * 3 + 2] = o[2];
        }
    }
}

extern "C" void kernel_launch(void* const* d_in, const int* in_sizes, int n_in,
                              void* d_out, int out_size, void* d_ws, size_t ws_size,
                              hipStream_t stream) {
    const float* x   = (const float*)d_in[0];
    float*       out = (float*)d_out;

    const int n_rows     = in_sizes[0] / 9;            // 4,000,000
    const int full_tiles = n_rows / TILE_ROWS;         // 3906
    const int rem        = n_rows - full_tiles * TILE_ROWS;
    const int grid       = full_tiles + (rem ? 1 : 0); // 3907

    hipLaunchKernelGGL(concat_layer_kernel, dim3(grid), dim3(BLOCK), 0, stream,
                       x, out, n_rows, full_tiles);
}